// Head_38963943309606
// MI455X (gfx1250) — compile-verified
//
#include <hip/hip_runtime.h>
#include <hip/hip_bf16.h>

typedef float v2f __attribute__((ext_vector_type(2)));
typedef float v8f __attribute__((ext_vector_type(8)));

#define D 128

__device__ __forceinline__ float lrelu(float x) { return x > 0.f ? x : 0.2f * x; }

// Order-preserving float <-> int encoding for atomicMax on floats.
__device__ __forceinline__ int fenc(float f) {
    int i = __float_as_int(f);
    return i >= 0 ? i : (int)(0x80000000u - (unsigned)i);
}
__device__ __forceinline__ float fdec(int e) {
    int i = e >= 0 ? e : (int)(0x80000000u - (unsigned)e);
    return __int_as_float(i);
}

// ---------------------------------------------------------------------------
// C[M x 128] = A[M x 128] @ B[128 x 128] via V_WMMA_F32_16X16X4_F32.
// One wave per 16x16 tile; 8 waves/block -> one block = 16 x 128 strip.
// ---------------------------------------------------------------------------
__global__ __launch_bounds__(256) void gemm128_wmma(const float* __restrict__ A,
                                                    const float* __restrict__ B,
                                                    float* __restrict__ C, int M) {
    const int wave = threadIdx.x >> 5;
    const int lane = threadIdx.x & 31;
    const int hh   = lane >> 4;      // half-wave select
    const int ln   = lane & 15;
    const int m0   = blockIdx.x << 4;
    const int n0   = wave << 4;
    if (m0 >= M) return;

    const float* __restrict__ arow = A + (size_t)(m0 + ln) * D;
    v8f c = {};
#pragma unroll 4
    for (int k0 = 0; k0 < D; k0 += 4) {
        const int kk = k0 + (hh << 1);
        v2f a, b;
        // A 16x4 fragment: lanes 0-15 K={k0,k0+1}, lanes 16-31 K={k0+2,k0+3}
        a.x = arow[kk];
        a.y = arow[kk + 1];
        // B 4x16 fragment: N = ln, same K striping as A
        b.x = B[(size_t)kk * D + n0 + ln];
        b.y = B[(size_t)(kk + 1) * D + n0 + ln];
        c = __builtin_amdgcn_wmma_f32_16x16x4_f32(false, a, false, b, (short)0, c,
                                                  false, false);
    }
    // C/D layout: VGPR i -> row (i + 8*hh), col = ln
    float* crow = C + (size_t)m0 * D + n0 + ln;
#pragma unroll
    for (int i = 0; i < 8; ++i) crow[(size_t)(i + (hh << 3)) * D] = c[i];
}

// ---------------------------------------------------------------------------
// Per-node: asv[n] = xp[n].a_src, adv[n] = xp[n].a_dst,
//           menc[n] = enc(leaky(asv+adv)) (self-loop seeds the max), deg[n]=1.
// One wave per node.
// ---------------------------------------------------------------------------
__global__ __launch_bounds__(256) void node_dots(const float* __restrict__ xp,
                                                 const float* __restrict__ a_src,
                                                 const float* __restrict__ a_dst,
                                                 float* __restrict__ asv,
                                                 float* __restrict__ adv,
                                                 int* __restrict__ menc,
                                                 float* __restrict__ deg, int N) {
    const int wave = threadIdx.x >> 5;
    const int lane = threadIdx.x & 31;
    const int n = blockIdx.x * 8 + wave;
    if (n >= N) return;
    const float4 v  = ((const float4*)(xp + (size_t)n * D))[lane];
    const float4 s4 = ((const float4*)a_src)[lane];
    const float4 d4 = ((const float4*)a_dst)[lane];
    float ds = v.x * s4.x + v.y * s4.y + v.z * s4.z + v.w * s4.w;
    float dd = v.x * d4.x + v.y * d4.y + v.z * d4.z + v.w * d4.w;
#pragma unroll
    for (int m = 16; m >= 1; m >>= 1) {
        ds += __shfl_xor(ds, m, 32);
        dd += __shfl_xor(dd, m, 32);
    }
    if (lane == 0) {
        asv[n] = ds;
        adv[n] = dd;
        menc[n] = fenc(lrelu(ds + dd));
        deg[n] = 1.0f;  // self-loop weight
    }
}

__global__ void zerok(float* __restrict__ p, int count) {
    const int i = blockIdx.x * blockDim.x + threadIdx.x;
    if (i < count) p[i] = 0.f;
}

// Pass: segment max of leaky(e) via encoded atomicMax; degree accumulation.
__global__ void gat_edge_max(const int* __restrict__ ei, const float* __restrict__ em,
                             const float* __restrict__ asv, const float* __restrict__ adv,
                             int* __restrict__ menc, float* __restrict__ deg, int E) {
    const int e = blockIdx.x * blockDim.x + threadIdx.x;
    if (e >= E) return;
    const int s = ei[e], t = ei[E + e];
    atomicMax(menc + t, fenc(lrelu(asv[s] + adv[t])));
    atomicAdd(deg + t, em[e]);
}

// Decode max, seed softmax denom with self-loop term, finalize dinv.
__global__ void gat_node_mid(const float* __restrict__ asv, const float* __restrict__ adv,
                             const int* __restrict__ menc, float* __restrict__ mval,
                             float* __restrict__ sden, float* __restrict__ dinv, int N) {
    const int n = blockIdx.x * blockDim.x + threadIdx.x;
    if (n >= N) return;
    const float m = fdec(menc[n]);
    mval[n] = m;
    sden[n] = expf(lrelu(asv[n] + adv[n]) - m);
    const float dg = dinv[n];  // currently holds deg
    dinv[n] = dg > 0.f ? 1.0f / sqrtf(fmaxf(dg, 1e-12f)) : 0.f;
}

// Pass: softmax denominator.
__global__ void gat_edge_sum(const int* __restrict__ ei, const float* __restrict__ asv,
                             const float* __restrict__ adv, const float* __restrict__ mval,
                             float* __restrict__ sden, int E) {
    const int e = blockIdx.x * blockDim.x + threadIdx.x;
    if (e >= E) return;
    const int s = ei[e], t = ei[E + e];
    atomicAdd(sden + t, expf(lrelu(asv[s] + adv[t]) - mval[t]));
}

// GAT aggregation: one wave per edge, float4 gather + 4 f32 atomics per lane.
__global__ __launch_bounds__(256) void gat_edge_agg(const int* __restrict__ ei,
                                                    const float* __restrict__ asv,
                                                    const float* __restrict__ adv,
                                                    const float* __restrict__ mval,
                                                    const float* __restrict__ sden,
                                                    const float* __restrict__ xp,
                                                    float* __restrict__ acc, int E) {
    const int w = (int)((blockIdx.x * blockDim.x + threadIdx.x) >> 5);
    const int lane = threadIdx.x & 31;
    if (w >= E) return;
    const int s = ei[w], t = ei[E + w];
    const float alpha = expf(lrelu(asv[s] + adv[t]) - mval[t]) / sden[t];
    const float4 v = ((const float4*)(xp + (size_t)s * D))[lane];
    float* dp = acc + (size_t)t * D + lane * 4;
    atomicAdd(dp + 0, alpha * v.x);
    atomicAdd(dp + 1, alpha * v.y);
    atomicAdd(dp + 2, alpha * v.z);
    atomicAdd(dp + 3, alpha * v.w);
}

// h = acc + alpha_self * xp + b ; cur = h ; out = h (JK running max init)
__global__ void gat_node_final(const float* __restrict__ xp, float* __restrict__ cur,
                               float* __restrict__ out, const float* __restrict__ asv,
                               const float* __restrict__ adv, const float* __restrict__ mval,
                               const float* __restrict__ sden, const float* __restrict__ bg,
                               int N) {
    const int idx = blockIdx.x * blockDim.x + threadIdx.x;
    if (idx >= N * D) return;
    const int n = idx >> 7, j = idx & (D - 1);
    const float aself = expf(lrelu(asv[n] + adv[n]) - mval[n]) / sden[n];
    const float h = cur[idx] + aself * xp[idx] + bg[j];
    cur[idx] = h;
    out[idx] = h;
}

// GCN aggregation: one wave per edge.
__global__ __launch_bounds__(256) void gcn_edge(const int* __restrict__ ei,
                                                const float* __restrict__ em,
                                                const float* __restrict__ dinv,
                                                const float* __restrict__ z,
                                                float* __restrict__ acc, int E) {
    const int w = (int)((blockIdx.x * blockDim.x + threadIdx.x) >> 5);
    const int lane = threadIdx.x & 31;
    if (w >= E) return;
    const int s = ei[w], t = ei[E + w];
    const float nrm = dinv[s] * em[w] * dinv[t];
    const float4 v = ((const float4*)(z + (size_t)s * D))[lane];
    float* dp = acc + (size_t)t * D + lane * 4;
    atomicAdd(dp + 0, nrm * v.x);
    atomicAdd(dp + 1, nrm * v.y);
    atomicAdd(dp + 2, nrm * v.z);
    atomicAdd(dp + 3, nrm * v.w);
}

// r = relu(acc + dinv^2 * z + b); cur = r; out = max(out, r)
__global__ void gcn_node(const float* __restrict__ z, float* __restrict__ cur,
                         float* __restrict__ out, const float* __restrict__ dinv,
                         const float* __restrict__ b, int N) {
    const int idx = blockIdx.x * blockDim.x + threadIdx.x;
    if (idx >= N * D) return;
    const int n = idx >> 7, j = idx & (D - 1);
    const float di = dinv[n];
    const float r = fmaxf(cur[idx] + di * di * z[idx] + b[j], 0.f);
    cur[idx] = r;
    out[idx] = fmaxf(out[idx], r);
}

extern "C" void kernel_launch(void* const* d_in, const int* in_sizes, int n_in,
                              void* d_out, int out_size, void* d_ws, size_t ws_size,
                              hipStream_t stream) {
    const float* x    = (const float*)d_in[0];
    const int*   ei   = (const int*)d_in[1];
    const float* em   = (const float*)d_in[2];
    const float* Wg   = (const float*)d_in[3];
    const float* av   = (const float*)d_in[4];
    const float* ad   = (const float*)d_in[5];
    const float* bg   = (const float*)d_in[6];
    const float* Wgcn = (const float*)d_in[7];
    const float* bgcn = (const float*)d_in[8];
    float* out = (float*)d_out;

    const int N = in_sizes[0] / D;      // 50000
    const int E = in_sizes[2];          // 800000
    const int NL = in_sizes[8] / D;     // 3

    auto au = [](size_t v) { return (v + 255) & ~(size_t)255; };
    char* p = (char*)d_ws;
    float* cur  = (float*)p; p += au((size_t)N * D * 4);
    float* z    = (float*)p; p += au((size_t)N * D * 4);
    float* asv  = (float*)p; p += au((size_t)N * 4);
    float* adv  = (float*)p; p += au((size_t)N * 4);
    float* mval = (float*)p; p += au((size_t)N * 4);
    float* sden = (float*)p; p += au((size_t)N * 4);
    float* dinv = (float*)p; p += au((size_t)N * 4);  // deg then 1/sqrt(deg)
    int*   menc = (int*)p;   p += au((size_t)N * 4);

    const int gemmB  = (N + 15) / 16;
    const int nodeWB = (N + 7) / 8;            // wave-per-node
    const int edgeTB = (E + 255) / 256;        // thread-per-edge
    const int edgeWB = (E + 7) / 8;            // wave-per-edge
    const int nd     = N * D;
    const int ndB    = (nd + 255) / 256;
    const int nB     = (N + 255) / 256;

    // ---- GAT ----
    gemm128_wmma<<<gemmB, 256, 0, stream>>>(x, Wg, z, N);               // z = x @ W_gat
    node_dots<<<nodeWB, 256, 0, stream>>>(z, av, ad, asv, adv, menc, dinv, N);
    zerok<<<ndB, 256, 0, stream>>>(cur, nd);
    gat_edge_max<<<edgeTB, 256, 0, stream>>>(ei, em, asv, adv, menc, dinv, E);
    gat_node_mid<<<nB, 256, 0, stream>>>(asv, adv, menc, mval, sden, dinv, N);
    gat_edge_sum<<<edgeTB, 256, 0, stream>>>(ei, asv, adv, mval, sden, E);
    gat_edge_agg<<<edgeWB, 256, 0, stream>>>(ei, asv, adv, mval, sden, z, cur, E);
    gat_node_final<<<ndB, 256, 0, stream>>>(z, cur, out, asv, adv, mval, sden, bg, N);

    // ---- GCN stack + JK max ----
    for (int l = 0; l < NL; ++l) {
        gemm128_wmma<<<gemmB, 256, 0, stream>>>(cur, Wgcn + (size_t)l * D * D, z, N);
        zerok<<<ndB, 256, 0, stream>>>(cur, nd);
        gcn_edge<<<edgeWB, 256, 0, stream>>>(ei, em, dinv, z, cur, E);
        gcn_node<<<ndB, 256, 0, stream>>>(z, cur, out, dinv, bgcn + (size_t)l * D, N);
    }
}